// GAT_2929167696331
// MI455X (gfx1250) — compile-verified
//
#include <hip/hip_runtime.h>
#include <hip/hip_bf16.h>

typedef __attribute__((ext_vector_type(2))) float v2f;
typedef __attribute__((ext_vector_type(8))) float v8f;

#define NNODES 50000
#define NEDGES 800000
#define ETOT   850000      // edges + self loops
#define NFREQ  513
#define HC     128         // H*C
#define NTILES 3125        // NNODES/16

// ---------------------------------------------------------------------------
// Small GEMM: Wcomb[513,128] = fb[513,128] @ Wg[128,128]
// ---------------------------------------------------------------------------
__global__ void k_wcomb(const float* __restrict__ fb, const float* __restrict__ Wg,
                        float* __restrict__ Wc) {
    int i = blockIdx.x * blockDim.x + threadIdx.x;
    if (i >= NFREQ * 128) return;
    int r = i >> 7, c = i & 127;
    float s = 0.f;
    #pragma unroll 8
    for (int k = 0; k < 128; ++k) s += fb[r * 128 + k] * Wg[k * 128 + c];
    Wc[i] = s;
}

// ---------------------------------------------------------------------------
// Main GEMM: h[N,128] = x[N,513] @ Wcomb[513,128], fp32 WMMA 16x16x4.
// Wcomb fully resident in LDS (513 rows padded to 516, stride 132).
// 4 waves/block; wave w == attention head w (32 cols). Epilogue fuses the
// att_src / att_dst per-row dot products via cross-lane reduction.
// Next grid-stride X tile is prefetched into cache (global_prefetch_b8).
// ---------------------------------------------------------------------------
__global__ void k_gemm_main(const float* __restrict__ x, const float* __restrict__ Wc,
                            const float* __restrict__ att_src, const float* __restrict__ att_dst,
                            float* __restrict__ h, float* __restrict__ a_s,
                            float* __restrict__ a_d) {
    extern __shared__ float sm[];
    float* Wl = sm;              // 516 * 132
    float* Xl = Wl + 516 * 132;  // 16 * 516
    const int tid = threadIdx.x, lane = tid & 31, wv = tid >> 5;
    // stage weights (once per block)
    for (int i = tid; i < NFREQ * 128; i += 128) {
        int r = i >> 7, c = i & 127;
        Wl[r * 132 + c] = Wc[i];
    }
    for (int i = tid; i < 3 * 128; i += 128) {           // zero K-pad rows 513..515
        int r = 513 + (i >> 7), c = i & 127;
        Wl[r * 132 + c] = 0.f;
    }
    const int m16 = lane & 15;
    const int koff = (lane >> 4) << 1;
    const float as0 = att_src[wv * 32 + m16],      as1 = att_src[wv * 32 + 16 + m16];
    const float ad0 = att_dst[wv * 32 + m16],      ad1 = att_dst[wv * 32 + 16 + m16];
    __syncthreads();

    for (int tile = blockIdx.x; tile < NTILES; tile += gridDim.x) {
        const int row0 = tile << 4;
        for (int i = tid; i < 16 * 516; i += 128) {
            int m = i / 516, k = i % 516;
            Xl[i] = (k < NFREQ) ? x[(row0 + m) * NFREQ + k] : 0.f;
        }
        // prefetch next grid-stride tile of x into cache hierarchy
        const int ntile = tile + gridDim.x;
        if (ntile < NTILES) {
            const float* nx = x + (size_t)ntile * 16 * NFREQ;
            for (int off = tid * 64; off < 16 * NFREQ; off += 128 * 64)
                __builtin_prefetch(nx + off, 0, 1);
        }
        __syncthreads();

        v8f c0 = {}, c1 = {};
        const int cb = wv * 32 + m16;
        for (int k0 = 0; k0 < 516; k0 += 4) {
            const int kr = k0 + koff;
            v2f a; a.x = Xl[m16 * 516 + kr];      a.y = Xl[m16 * 516 + kr + 1];
            v2f b0; b0.x = Wl[kr * 132 + cb];      b0.y = Wl[(kr + 1) * 132 + cb];
            v2f b1; b1.x = Wl[kr * 132 + cb + 16]; b1.y = Wl[(kr + 1) * 132 + cb + 16];
            c0 = __builtin_amdgcn_wmma_f32_16x16x4_f32(false, a, false, b0, (short)0, c0, false, false);
            c1 = __builtin_amdgcn_wmma_f32_16x16x4_f32(false, a, false, b1, (short)0, c1, false, false);
        }

        // store h tile + fused attention-coefficient dot products
        #pragma unroll
        for (int i = 0; i < 8; ++i) {
            const int row = row0 + i + ((lane >> 4) << 3);
            h[row * HC + wv * 32 + m16]      = c0[i];
            h[row * HC + wv * 32 + 16 + m16] = c1[i];
            float sv = c0[i] * as0 + c1[i] * as1;
            float dv = c0[i] * ad0 + c1[i] * ad1;
            #pragma unroll
            for (int m = 1; m < 16; m <<= 1) {
                sv += __shfl_xor(sv, m, 32);
                dv += __shfl_xor(dv, m, 32);
            }
            if (m16 == 0) { a_s[row * 4 + wv] = sv; a_d[row * 4 + wv] = dv; }
        }
        __syncthreads();
    }
}

// ---------------------------------------------------------------------------
// Edge attention: monotone float->uint key for atomic segment-max
// ---------------------------------------------------------------------------
__device__ __forceinline__ unsigned fkey(float f) {
    unsigned u = __float_as_uint(f);
    return (u & 0x80000000u) ? ~u : (u | 0x80000000u);
}
__device__ __forceinline__ float funkey(unsigned k) {
    return __uint_as_float((k & 0x80000000u) ? (k & 0x7fffffffu) : ~k);
}
__device__ __forceinline__ void edge_nodes(const int* ei, int e, int& s, int& d) {
    if (e < NEDGES) { s = ei[e]; d = ei[NEDGES + e]; }
    else            { s = d = e - NEDGES; }          // self loops
}

__global__ void k_edge_max(const int* __restrict__ ei, const float* __restrict__ as,
                           const float* __restrict__ ad, unsigned* __restrict__ mkey) {
    int gid = blockIdx.x * blockDim.x + threadIdx.x;
    if (gid >= ETOT * 4) return;
    int e = gid >> 2, hh = gid & 3, s, d;
    edge_nodes(ei, e, s, d);
    float v = as[s * 4 + hh] + ad[d * 4 + hh];
    v = (v > 0.f) ? v : 0.2f * v;
    atomicMax(&mkey[d * 4 + hh], fkey(v));
}

// pass 2: numerator per (edge, head) stored, denominator accumulated
__global__ void k_edge_denom(const int* __restrict__ ei, const float* __restrict__ as,
                             const float* __restrict__ ad, const unsigned* __restrict__ mkey,
                             float* __restrict__ denom, float* __restrict__ exn) {
    int gid = blockIdx.x * blockDim.x + threadIdx.x;
    if (gid >= ETOT * 4) return;
    int e = gid >> 2, hh = gid & 3, s, d;
    edge_nodes(ei, e, s, d);
    float v = as[s * 4 + hh] + ad[d * 4 + hh];
    v = (v > 0.f) ? v : 0.2f * v;
    float ex = __expf(v - funkey(mkey[d * 4 + hh]));
    exn[gid] = ex;
    atomicAdd(&denom[d * 4 + hh], ex);
}

// pass 3: message scatter, one thread per (edge, channel)
__global__ void k_edge_scatter(const int* __restrict__ ei, const float* __restrict__ exn,
                               const float* __restrict__ denom, const float* __restrict__ h,
                               float* __restrict__ acc) {
    int gid = blockIdx.x * blockDim.x + threadIdx.x;
    if (gid >= ETOT * HC) return;
    int e = gid >> 7, c = gid & 127, hh = c >> 5, s, d;
    edge_nodes(ei, e, s, d);
    float alpha = exn[e * 4 + hh] / denom[d * 4 + hh];
    atomicAdd(&acc[d * HC + c], h[s * HC + c] * alpha);
}

// bias + ELU, in place on acc
__global__ void k_bias_elu(float* __restrict__ acc, const float* __restrict__ bias_g) {
    int i = blockIdx.x * blockDim.x + threadIdx.x;
    if (i >= NNODES * HC) return;
    float v = acc[i] + bias_g[i & 127];
    acc[i] = (v > 0.f) ? v : (__expf(v) - 1.f);
}

// ---------------------------------------------------------------------------
// Generic fp32-WMMA GEMM + bias + leaky_relu(0.01): out[N,NOUT] = A[N,K] @ W
// Weights + bias LDS-resident; 4 waves/block, each wave owns CT 16-col tiles
// across TWO 16-row M tiles (B fragments reused 2x per LDS read).
// ---------------------------------------------------------------------------
template <int K, int NOUT, int CT>
__global__ void k_gemm_lrelu(const float* __restrict__ A, const float* __restrict__ W,
                             const float* __restrict__ bias, float* __restrict__ out) {
    extern __shared__ float sm[];
    constexpr int WS = NOUT + 4;  // LDS W stride (mod 64 == 4: conflict-free)
    constexpr int AS = K + 4;     // LDS A stride
    constexpr int RT = 2;         // row tiles per block (32 rows)
    float* Wl = sm;               // K * WS
    float* Al = Wl + K * WS;      // 32 * AS
    float* Bl = Al + 32 * AS;     // NOUT
    const int tid = threadIdx.x, lane = tid & 31, wv = tid >> 5;
    for (int i = tid; i < K * NOUT; i += 128) {
        int r = i / NOUT, c = i % NOUT;
        Wl[r * WS + c] = W[i];
    }
    for (int i = tid; i < NOUT; i += 128) Bl[i] = bias[i];
    const int m16 = lane & 15;
    const int koff = (lane >> 4) << 1;
    const int ntiles32 = (NNODES + 31) / 32;
    __syncthreads();

    for (int tile = blockIdx.x; tile < ntiles32; tile += gridDim.x) {
        const int row0 = tile << 5;
        for (int i = tid; i < 32 * K; i += 128) {
            int m = i / K, k = i % K;
            int r = row0 + m;
            Al[m * AS + k] = (r < NNODES) ? A[r * K + k] : 0.f;
        }
        __syncthreads();

        v8f accv[RT][CT] = {};
        for (int k0 = 0; k0 < K; k0 += 4) {
            const int kr = k0 + koff;
            v2f a[RT];
            #pragma unroll
            for (int rt = 0; rt < RT; ++rt) {
                a[rt].x = Al[(rt * 16 + m16) * AS + kr];
                a[rt].y = Al[(rt * 16 + m16) * AS + kr + 1];
            }
            #pragma unroll
            for (int t = 0; t < CT; ++t) {
                const int cbt = wv * (CT * 16) + t * 16 + m16;
                v2f b; b.x = Wl[kr * WS + cbt]; b.y = Wl[(kr + 1) * WS + cbt];
                #pragma unroll
                for (int rt = 0; rt < RT; ++rt)
                    accv[rt][t] = __builtin_amdgcn_wmma_f32_16x16x4_f32(
                        false, a[rt], false, b, (short)0, accv[rt][t], false, false);
            }
        }
        #pragma unroll
        for (int t = 0; t < CT; ++t) {
            const int col = wv * (CT * 16) + t * 16 + m16;
            const float bv = Bl[col];
            #pragma unroll
            for (int rt = 0; rt < RT; ++rt) {
                #pragma unroll
                for (int i = 0; i < 8; ++i) {
                    const int row = row0 + rt * 16 + i + ((lane >> 4) << 3);
                    if (row < NNODES) {
                        float v = accv[rt][t][i] + bv;
                        out[row * NOUT + col] = (v > 0.f) ? v : 0.01f * v;
                    }
                }
            }
        }
        __syncthreads();
    }
}

// ---------------------------------------------------------------------------
// Final layer (128 -> 10) + leaky_relu + softmax, one thread per node
// ---------------------------------------------------------------------------
__global__ void k_final(const float* __restrict__ act2, const float* __restrict__ W3,
                        const float* __restrict__ b3, float* __restrict__ out) {
    int n = blockIdx.x * blockDim.x + threadIdx.x;
    if (n >= NNODES) return;
    float z[10];
    #pragma unroll
    for (int j = 0; j < 10; ++j) z[j] = b3[j];
    const float* a = act2 + n * 128;
    for (int k = 0; k < 128; ++k) {
        float av = a[k];
        #pragma unroll
        for (int j = 0; j < 10; ++j) z[j] += av * W3[k * 10 + j];
    }
    float mx = -3.4e38f;
    #pragma unroll
    for (int j = 0; j < 10; ++j) {
        z[j] = (z[j] > 0.f) ? z[j] : 0.01f * z[j];
        mx = fmaxf(mx, z[j]);
    }
    float se = 0.f;
    #pragma unroll
    for (int j = 0; j < 10; ++j) { z[j] = __expf(z[j] - mx); se += z[j]; }
    float inv = 1.f / se;
    #pragma unroll
    for (int j = 0; j < 10; ++j) out[n * 10 + j] = z[j] * inv;
}

// ---------------------------------------------------------------------------
extern "C" void kernel_launch(void* const* d_in, const int* in_sizes, int n_in,
                              void* d_out, int out_size, void* d_ws, size_t ws_size,
                              hipStream_t stream) {
    const float* x       = (const float*)d_in[0];
    const int*   ei      = (const int*)d_in[1];
    const float* fb      = (const float*)d_in[2];
    const float* Wg      = (const float*)d_in[3];
    const float* bias_g  = (const float*)d_in[4];
    const float* att_src = (const float*)d_in[5];
    const float* att_dst = (const float*)d_in[6];
    const float* W1      = (const float*)d_in[7];
    const float* b1      = (const float*)d_in[8];
    const float* W2      = (const float*)d_in[9];
    const float* b2      = (const float*)d_in[10];
    const float* W3      = (const float*)d_in[11];
    const float* b3      = (const float*)d_in[12];
    float* out = (float*)d_out;

    float*    ws    = (float*)d_ws;
    float*    Wc    = ws;                      // 513*128          = 65,664
    float*    h     = Wc + 65664;              // N*128            = 6,400,000
    float*    a_s   = h + 6400000;             // N*4              = 200,000
    float*    a_d   = a_s + 200000;            // N*4
    unsigned* mkey  = (unsigned*)(a_d + 200000);   // N*4
    float*    denom = (float*)(mkey + 200000);     // N*4
    float*    acc   = denom + 200000;          // N*128 (act0 in place)
    float*    act1  = acc + 6400000;           // N*256
    float*    exn   = act1 + 12800000;         // ETOT*4           = 3,400,000
    float*    act2  = h;                       // reuse h region

    hipMemsetAsync(mkey, 0, (size_t)NNODES * 4 * sizeof(unsigned), stream);
    hipMemsetAsync(denom, 0, (size_t)NNODES * 4 * sizeof(float), stream);
    hipMemsetAsync(acc, 0, (size_t)NNODES * HC * sizeof(float), stream);

    k_wcomb<<<(NFREQ * 128 + 255) / 256, 256, 0, stream>>>(fb, Wg, Wc);

    const size_t lds_main = (size_t)(516 * 132 + 16 * 516) * 4;          // 305,472 B
    k_gemm_main<<<640, 128, lds_main, stream>>>(x, Wc, att_src, att_dst, h, a_s, a_d);

    k_edge_max<<<(ETOT * 4 + 255) / 256, 256, 0, stream>>>(ei, a_s, a_d, mkey);
    k_edge_denom<<<(ETOT * 4 + 255) / 256, 256, 0, stream>>>(ei, a_s, a_d, mkey, denom, exn);
    k_edge_scatter<<<(ETOT * HC + 255) / 256, 256, 0, stream>>>(ei, exn, denom, h, acc);
    k_bias_elu<<<(NNODES * HC + 255) / 256, 256, 0, stream>>>(acc, bias_g);

    const size_t lds_g1 = (size_t)(128 * 260 + 32 * 132 + 256) * 4;      // 151,040 B
    k_gemm_lrelu<128, 256, 4><<<640, 128, lds_g1, stream>>>(acc, W1, b1, act1);
    const size_t lds_g2 = (size_t)(256 * 132 + 32 * 260 + 128) * 4;      // 168,960 B
    k_gemm_lrelu<256, 128, 2><<<640, 128, lds_g2, stream>>>(act1, W2, b2, act2);

    k_final<<<(NNODES + 255) / 256, 256, 0, stream>>>(act2, W3, b3, out);
}